// DeepFM_25366076850614
// MI455X (gfx1250) — compile-verified
//
#include <hip/hip_runtime.h>
#include <hip/hip_bf16.h>

typedef __attribute__((ext_vector_type(2))) float v2f;
typedef __attribute__((ext_vector_type(8))) float v8f;

#define B_ 16384
#define F_ 40
#define ND_ 13
#define E_ 16
#define V_ 100000
#define D0_ 640
#define H1_ 512
#define H2_ 256

// ---------------------------------------------------------------------------
// Kernel 1: embeddings + FM first/second order. One wave per sample.
// lane -> (field half, e): f = f0 + lane/16, e = lane%16. Sparse gather of a
// 16-float (64B) embedding row is served by 16 consecutive lanes -> coalesced.
// ---------------------------------------------------------------------------
__global__ __launch_bounds__(256) void deepfm_embed_fm(
    const int* __restrict__ Xi, const float* __restrict__ Xv,
    const float* __restrict__ bias,
    const float* __restrict__ dw1, const float* __restrict__ db1,
    const float* __restrict__ e1,
    const float* __restrict__ dw2, const float* __restrict__ db2,
    const float* __restrict__ e2,
    float* __restrict__ so_flat, float* __restrict__ partial)
{
    int wid  = threadIdx.x >> 5;
    int lane = threadIdx.x & 31;
    int b = blockIdx.x * 8 + wid;
    if (b >= B_) return;

    int e     = lane & 15;
    int fhalf = lane >> 4;       // 0 or 1

    float fm1 = 0.f;             // sum of first-order embed elements (this lane)
    float s_part = 0.f;          // partial s_e over this lane's fields
    float sq_part = 0.f;         // partial sum of so^2

    for (int f0 = 0; f0 < F_; f0 += 2) {
        int f = f0 + fhalf;
        float xv = Xv[b * F_ + f];
        int   xi = Xi[b * F_ + f];
        float v1, v2;
        if (f < ND_) {
            float xd = (float)xi;
            v1 = xd * dw1[f * E_ + e] + db1[f * E_ + e];
            v2 = xd * dw2[f * E_ + e] + db2[f * E_ + e];
        } else {
            size_t off = (size_t)(f - ND_) * ((size_t)V_ * E_) + (size_t)xi * E_ + e;
            v1 = e1[off];
            v2 = e2[off];
        }
        v1 *= xv;
        v2 *= xv;
        fm1     += v1;
        s_part  += v2;
        sq_part += v2 * v2;
        so_flat[(size_t)b * D0_ + f * E_ + e] = v2;
    }
    // s_e = sum over all fields for this e (combine the two field-halves)
    float s_e = s_part + __shfl_xor(s_part, 16, 32);
    // fm contribution: fm1 - 0.5*sq (per lane) + 0.5*s_e^2 (count each e once)
    float contrib = fm1 - 0.5f * sq_part + ((fhalf == 0) ? 0.5f * s_e * s_e : 0.f);
    #pragma unroll
    for (int m = 16; m >= 1; m >>= 1) contrib += __shfl_xor(contrib, m, 32);
    if (lane == 0) partial[b] = contrib + bias[b];
}

// ---------------------------------------------------------------------------
// Kernel 2/5: fp32 WMMA GEMM  C[M,N] = A @ (scale_row .* Bw) + bias_row
// Block: 256 threads (8 waves). Block tile 128x64; each wave owns a 16-row
// strip with FOUR 16x16 accumulators (A-fragment reused 4x).
//
// LDS layouts (both double-buffered, both conflict-free, both fragments are
// adjacent (k,k+1) pairs -> single ds_load_b64, no register shuffles):
//   As: row-major, row stride 20 floats (80B: 16B-aligned sub-rows) -> staged
//       raw with gfx1250 global_load_async_to_lds_b128 (ASYNCcnt).
//   Bs: TRANSPOSED [col][k], col stride 18 floats -> staged synchronously;
//       the BN1 fold (A.*a @ W2 == A @ diag(a)W2) is applied here as a
//       per-k scalar scale on B rows.
// ---------------------------------------------------------------------------
#define TMm 128
#define TN  64
#define TK  16
#define ASTRIDE 20
#define BSTRIDE 18

__device__ __forceinline__ void async_b128_to_lds(const float* g, const float* lds)
{
    // VDST = LDS byte offset (low 32 bits of generic LDS pointer),
    // VADDR = 64-bit global address. Tracked by ASYNCcnt.
    unsigned int lo = (unsigned int)(unsigned long long)lds;
    asm volatile("global_load_async_to_lds_b128 %0, %1, off"
                 :: "v"(lo), "v"(g) : "memory");
}

template<bool HAS_SCALE>
__global__ __launch_bounds__(256) void deepfm_gemm_wmma(
    const float* __restrict__ A, const float* __restrict__ Bw,
    const float* __restrict__ bias, const float* __restrict__ scale,
    float* __restrict__ C, int M, int K, int N)
{
    __shared__ __align__(16) float As[2][TMm * ASTRIDE];
    __shared__ __align__(16) float Bs[2][TN * BSTRIDE];

    int tid  = threadIdx.x;
    int wid  = tid >> 5;
    int lane = tid & 31;

    int tilesN  = N / TN;
    int rowBase = (blockIdx.x / tilesN) * TMm;
    int colBase = (blockIdx.x % tilesN) * TN;

    int m_frag = lane & 15;          // A row / B col within fragment
    int ko     = (lane >> 4) * 2;    // K pair selected by lane[4] (ISA layout)
    int wrow   = wid * 16;           // wave's 16-row strip in the block tile

    // per-thread staging pointers (advance by TK each chunk; no per-iter mul)
    int aRow = tid >> 2;             // 0..63 (+64 for second half)
    int aCg  = (tid & 3) * 4;
    const float* gA0 = A + (size_t)(rowBase + aRow) * K + aCg;
    const float* gA1 = gA0 + (size_t)64 * K;
    int bRow = tid >> 4;             // 0..15 (k within chunk)
    int bCg  = (tid & 15) * 4;       // 4-col group
    const float* gB  = Bw + (size_t)bRow * N + colBase + bCg;
    const float* gS  = HAS_SCALE ? (scale + bRow) : nullptr;

    v8f acc[4] = {{}, {}, {}, {}};

    auto stageA = [&](int buf) {     // async, raw
        async_b128_to_lds(gA0, &As[buf][aRow * ASTRIDE + aCg]);
        async_b128_to_lds(gA1, &As[buf][(aRow + 64) * ASTRIDE + aCg]);
    };
    auto stageB = [&](int buf) {     // sync, transposed scatter, per-k scale
        float4 v = *(const float4*)gB;
        if (HAS_SCALE) {
            float sc = *gS;
            v.x *= sc; v.y *= sc; v.z *= sc; v.w *= sc;
        }
        float* d = &Bs[buf][bCg * BSTRIDE + bRow];
        d[0 * BSTRIDE] = v.x;
        d[1 * BSTRIDE] = v.y;
        d[2 * BSTRIDE] = v.z;
        d[3 * BSTRIDE] = v.w;
    };
    auto advance = [&]() {
        gA0 += TK; gA1 += TK; gB += (size_t)TK * N;
        if (HAS_SCALE) gS += TK;
    };

    stageA(0);
    stageB(0);

    int nChunks = K / TK;
    for (int i = 0; i < nChunks; i++) {
        int buf = i & 1;
        // my async A-stage (B ds-stores handled by compiler) must land
        asm volatile("s_wait_asynccnt 0" ::: "memory");
        __syncthreads();
        if (i + 1 < nChunks) {
            advance();
            stageA(buf ^ 1);
            stageB(buf ^ 1);
        }
        #pragma unroll
        for (int kk = 0; kk < TK; kk += 4) {
            v2f a = *(const v2f*)&As[buf][(wrow + m_frag) * ASTRIDE + kk + ko];
            #pragma unroll
            for (int j = 0; j < 4; j++) {
                v2f b = *(const v2f*)&Bs[buf][(j * 16 + m_frag) * BSTRIDE + kk + ko];
                acc[j] = __builtin_amdgcn_wmma_f32_16x16x4_f32(
                             false, a, false, b, (short)0, acc[j], false, false);
            }
        }
        __syncthreads();   // all reads of buf done before it is restaged
    }

    // --- epilogue: D layout = VGPR r -> row r + 8*(lane>=16), col lane%16 ---
    int nIdx = lane & 15;
    int rOff = (lane >> 4) * 8;
    #pragma unroll
    for (int j = 0; j < 4; j++) {
        int col = colBase + j * 16 + nIdx;
        float bv = bias[col];
        #pragma unroll
        for (int r = 0; r < 8; r++) {
            int row = rowBase + wrow + rOff + r;
            C[(size_t)row * N + col] = acc[j][r] + bv;
        }
    }
}

// ---------------------------------------------------------------------------
// Kernel 3/6: batchnorm column statistics -> affine fold  a = g*rsqrt(var+eps),
// c = beta - mu*a.  Block = 64 columns x 4 row-threads (coalesced rows).
// ---------------------------------------------------------------------------
__global__ __launch_bounds__(256) void deepfm_bn_reduce(
    const float* __restrict__ Z, const float* __restrict__ g,
    const float* __restrict__ bt, float* __restrict__ a, float* __restrict__ c,
    int M, int N)
{
    int ccol = threadIdx.x & 63;
    int rrow = threadIdx.x >> 6;        // 0..3
    int col  = blockIdx.x * 64 + ccol;

    float sum = 0.f, sq = 0.f;
    for (int r = rrow; r < M; r += 4) {
        float v = Z[(size_t)r * N + col];
        sum += v;
        sq  += v * v;
    }
    __shared__ float ssum[256];
    __shared__ float ssq[256];
    ssum[threadIdx.x] = sum;
    ssq[threadIdx.x]  = sq;
    __syncthreads();
    if (rrow == 0) {
        sum = ssum[ccol] + ssum[64 + ccol] + ssum[128 + ccol] + ssum[192 + ccol];
        sq  = ssq[ccol]  + ssq[64 + ccol]  + ssq[128 + ccol]  + ssq[192 + ccol];
        float inv = 1.f / (float)M;
        float mu  = sum * inv;
        float var = sq * inv - mu * mu;
        float av  = g[col] * rsqrtf(var + 1e-5f);
        a[col] = av;
        c[col] = bt[col] - mu * av;
    }
}

// ---------------------------------------------------------------------------
// Kernel 4: cvec[j] = b2[j] + sum_i c1[i] * W2[i,j]   (fold BN1 shift into
// GEMM2's bias row). One block, 256 threads (j), loop over i.
// ---------------------------------------------------------------------------
__global__ __launch_bounds__(256) void deepfm_cvec(
    const float* __restrict__ c1, const float* __restrict__ W2,
    const float* __restrict__ b2, float* __restrict__ cvec, int K, int N)
{
    int j = threadIdx.x;
    if (j >= N) return;
    float acc = b2[j];
    for (int i = 0; i < K; i++)
        acc += c1[i] * W2[(size_t)i * N + j];
    cvec[j] = acc;
}

// ---------------------------------------------------------------------------
// Kernel 7: out[b] = partial[b] + sum_j (Z2[b,j]*a2[j] + c2[j]).
// One wave per row, lanes stride columns (coalesced), wave shuffle reduce.
// ---------------------------------------------------------------------------
__global__ __launch_bounds__(256) void deepfm_final(
    const float* __restrict__ Z2, const float* __restrict__ a2,
    const float* __restrict__ c2, const float* __restrict__ partial,
    float* __restrict__ out, int M, int N)
{
    int wid  = threadIdx.x >> 5;
    int lane = threadIdx.x & 31;
    int row  = blockIdx.x * 8 + wid;
    if (row >= M) return;
    float acc = 0.f;
    for (int j = lane; j < N; j += 32)
        acc += Z2[(size_t)row * N + j] * a2[j] + c2[j];
    #pragma unroll
    for (int m = 16; m >= 1; m >>= 1) acc += __shfl_xor(acc, m, 32);
    if (lane == 0) out[row] = acc + partial[row];
}

// ---------------------------------------------------------------------------
extern "C" void kernel_launch(void* const* d_in, const int* in_sizes, int n_in,
                              void* d_out, int out_size, void* d_ws, size_t ws_size,
                              hipStream_t stream)
{
    const int*   Xi   = (const int*)  d_in[0];
    const float* Xv   = (const float*)d_in[1];
    const float* bias = (const float*)d_in[2];
    const float* dw1  = (const float*)d_in[3];
    const float* db1  = (const float*)d_in[4];
    const float* e1   = (const float*)d_in[5];
    const float* dw2  = (const float*)d_in[6];
    const float* db2  = (const float*)d_in[7];
    const float* e2   = (const float*)d_in[8];
    const float* W1   = (const float*)d_in[9];
    const float* b1   = (const float*)d_in[10];
    const float* g1   = (const float*)d_in[11];
    const float* bt1  = (const float*)d_in[12];
    const float* W2   = (const float*)d_in[13];
    const float* b2   = (const float*)d_in[14];
    const float* g2   = (const float*)d_in[15];
    const float* bt2  = (const float*)d_in[16];
    float* out = (float*)d_out;

    // workspace carve-up (all 256B aligned)
    char* ws = (char*)d_ws;
    size_t off = 0;
    auto carve = [&](size_t bytes) {
        char* p = ws + off;
        off += (bytes + 255) & ~(size_t)255;
        return p;
    };
    float* so      = (float*)carve((size_t)B_ * D0_ * 4);  // 41.9 MB
    float* Z1      = (float*)carve((size_t)B_ * H1_ * 4);  // 33.5 MB
    float* Z2      = (float*)carve((size_t)B_ * H2_ * 4);  // 16.8 MB
    float* partial = (float*)carve((size_t)B_ * 4);
    float* a1      = (float*)carve(H1_ * 4);
    float* c1      = (float*)carve(H1_ * 4);
    float* a2      = (float*)carve(H2_ * 4);
    float* c2      = (float*)carve(H2_ * 4);
    float* cvec    = (float*)carve(H2_ * 4);
    (void)ws_size;

    // 1) embeddings + FM terms, produce so (GEMM1 A-matrix) and partial out
    deepfm_embed_fm<<<B_ / 8, 256, 0, stream>>>(
        Xi, Xv, bias, dw1, db1, e1, dw2, db2, e2, so, partial);

    // 2) Z1 = so @ W1 + b1          (M=16384, K=640, N=512)
    deepfm_gemm_wmma<false><<<(B_ / TMm) * (H1_ / TN), 256, 0, stream>>>(
        so, W1, b1, nullptr, Z1, B_, D0_, H1_);

    // 3) BN1 statistics -> a1, c1
    deepfm_bn_reduce<<<H1_ / 64, 256, 0, stream>>>(Z1, g1, bt1, a1, c1, B_, H1_);

    // 4) cvec = c1 @ W2 + b2
    deepfm_cvec<<<1, 256, 0, stream>>>(c1, W2, b2, cvec, H1_, H2_);

    // 5) Z2 = Z1 @ (a1.*W2) + cvec  (M=16384, K=512, N=256), BN1 folded in
    deepfm_gemm_wmma<true><<<(B_ / TMm) * (H2_ / TN), 256, 0, stream>>>(
        Z1, W2, cvec, a1, Z2, B_, H1_, H2_);

    // 6) BN2 statistics -> a2, c2
    deepfm_bn_reduce<<<H2_ / 64, 256, 0, stream>>>(Z2, g2, bt2, a2, c2, B_, H2_);

    // 7) out = partial + rowsum(Z2*a2 + c2)
    deepfm_final<<<B_ / 8, 256, 0, stream>>>(Z2, a2, c2, partial, out, B_, H2_);
}